// CannyFilter_50964081935593
// MI455X (gfx1250) — compile-verified
//
#include <hip/hip_runtime.h>
#include <cstdint>

// ---------------------------------------------------------------------------
// Fused Canny (blur + sobel + grad accumulation + octant NMS) for gfx1250.
// - Tile staging: CDNA5 Tensor Data Mover (tensor_load_to_lds, TENSORcnt) for
//   all tiles except left/top image borders (those use async global->LDS +
//   explicit zero-pad).
// - Separable 5-tap blurs: V_WMMA_F32_16X16X4_F32 against a constant banded
//   weight matrix (full fp32 precision); offloads ~30% of VALU work to the
//   matrix pipe in a VALU-issue-bound kernel.
// - Sobel/magnitude/NMS: VALU with per-thread register accumulators.
// ---------------------------------------------------------------------------

#define TILE   64          // output tile (64x64 per workgroup)
#define APRON  4           // total stencil radius: blur(2) + sobel(1) + nms(1)
#define IN_W   (TILE + 2*APRON)   // 72 : raw input tile
#define HB_W   (TILE + 4)         // 68 : h-blurred width (radius 2 kept in x)
#define VB_W   (TILE + 4)         // 68 : v-blurred tile (radius 2)
#define GR_W   (TILE + 2)         // 66 : grad_mag tile (radius 1 for NMS)
#define IMG    1024
#define NTHR   256
#define ROWS_PER_WAVE (IN_W / 8)  // 9 rows per wave, 8 waves

// Gaussian (reference does NOT normalize): exp(-[2,0.5,0,0.5,2])
#define G0 0.13533528f
#define G1 0.60653067f
// tan(22.5 deg)
#define TAN22 0.41421356f

typedef unsigned int v4u __attribute__((ext_vector_type(4)));
typedef int          v8i __attribute__((ext_vector_type(8)));
typedef int          v4i __attribute__((ext_vector_type(4)));
typedef float        v2f __attribute__((ext_vector_type(2)));
typedef float        v8f __attribute__((ext_vector_type(8)));

#if __has_builtin(__builtin_amdgcn_tensor_load_to_lds)
#define HAVE_TDM 1
#else
#define HAVE_TDM 0
#endif

#if __has_builtin(__builtin_amdgcn_wmma_f32_16x16x4_f32)
#define HAVE_WMMA4 1
#else
#define HAVE_WMMA4 0
#endif

// ---- CDNA5 async global->LDS copy (tracked by ASYNCcnt) -------------------
__device__ __forceinline__ void async_ld_b32(const float* g, const float* lds) {
  unsigned l = (unsigned)(uintptr_t)lds;   // low 32 bits of generic = LDS addr
  asm volatile("global_load_async_to_lds_b32 %0, %1, off"
               :: "v"(l), "v"(g) : "memory");
}
__device__ __forceinline__ void async_ld_b128(const float* g, const float* lds) {
  unsigned l = (unsigned)(uintptr_t)lds;
  asm volatile("global_load_async_to_lds_b128 %0, %1, off"
               :: "v"(l), "v"(g) : "memory");
}
__device__ __forceinline__ void wait_async0() {
#if __has_builtin(__builtin_amdgcn_s_wait_asynccnt)
  __builtin_amdgcn_s_wait_asynccnt(0);
#else
  asm volatile("s_wait_asynccnt 0" ::: "memory");
#endif
}
__device__ __forceinline__ void wait_tensor0() {
#if __has_builtin(__builtin_amdgcn_s_wait_tensorcnt)
  __builtin_amdgcn_s_wait_tensorcnt(0);
#else
  asm volatile("s_wait_tensorcnt 0" ::: "memory");
#endif
}

#if HAVE_TDM
// ---- Tensor Data Mover: 2-D tile load, one descriptor per wave ------------
// D# packing per CDNA5 ISA ch.8: group0 {count, lds_addr, global_addr, type=2},
// group1 {data_size=4B, tensor_dim0/1 = remaining extents from tile origin
// (hardware zero-fills past right/bottom image edges), tile dims, dim0 stride}.
// Groups 2/3 zero => 2-D tensor.
__device__ __forceinline__ void tdm_load_rows(const float* gaddr, unsigned lds,
                                              unsigned rows, unsigned rem_x,
                                              unsigned rem_y) {
  unsigned long long ga = (unsigned long long)(uintptr_t)gaddr;
  v4u g0;
  g0.x = 1u;                                         // count=1, no gather
  g0.y = lds;                                        // lds_addr [63:32]
  g0.z = (unsigned)ga;                               // global_addr lo
  g0.w = (unsigned)((ga >> 32) & 0x01FFFFFFu)        // global_addr [56:32]
       | (2u << 30);                                 // type=2 ("image")
  v8i g1;
  g1[0] = 0x00020000;                                // data_size=4B, mask=0
  g1[1] = (int)((rem_x & 0xFFFFu) << 16);            // tensor_dim0 lo16 @bit48
  g1[2] = (int)(((rem_x >> 16) & 0xFFFFu) | ((rem_y & 0xFFFFu) << 16));
  g1[3] = (int)(((rem_y >> 16) & 0xFFFFu) | ((unsigned)IN_W << 16)); // tile_dim0=72
  g1[4] = (int)(rows & 0xFFFFu);                     // tile_dim1, tile_dim2=0
  g1[5] = IMG;                                       // tensor_dim0_stride lo32
  g1[6] = 0;                                         // stride hi, dim1_stride lo
  g1[7] = 0;
  v4i z4 = {0, 0, 0, 0};
#if defined(__clang_major__) && __clang_major__ >= 23
  v8i z8 = {0, 0, 0, 0, 0, 0, 0, 0};
  __builtin_amdgcn_tensor_load_to_lds(g0, g1, z4, z4, z8, 0);
#else
  __builtin_amdgcn_tensor_load_to_lds(g0, g1, z4, z4, 0);
#endif
}
#endif

// Banded 5-tap weight: W[d] = g[d] for d in [0,4], else 0.
__device__ __forceinline__ float band_w(int d) {
  float v = 0.0f;
  v = (d == 0 || d == 4) ? G0 : v;
  v = (d == 1 || d == 3) ? G1 : v;
  v = (d == 2) ? 1.0f : v;
  return v;
}

__global__ __launch_bounds__(NTHR)
void canny_fused_kernel(const float* __restrict__ img, float* __restrict__ out) {
  __shared__ float sIn[IN_W * IN_W];   // 5184 f: raw tile; reused as Vb (68x68)
  __shared__ float sHb[IN_W * HB_W];   // 72x68 : horizontally blurred
  __shared__ float sGr[GR_W * GR_W];   // 66x66 : accumulated grad magnitude

  const int tid = threadIdx.x;
  const int x0  = blockIdx.x * TILE;
  const int y0  = blockIdx.y * TILE;
  const int b   = blockIdx.z;

#if HAVE_TDM
  const bool fast = (x0 >= APRON) && (y0 >= APRON);
  const unsigned wave = (unsigned)__builtin_amdgcn_readfirstlane(tid) >> 5;
#else
  const bool fast = (x0 >= APRON) && (x0 + TILE + APRON <= IMG) &&
                    (y0 >= APRON) && (y0 + TILE + APRON <= IMG);
#endif

#if HAVE_WMMA4
  // Constant banded-weight fragments (5 K-chunks of 4), built once.
  // A 16x4 f32 layout (documented): lane l -> M=l&15, Kbase=(l>>4)*2.
  // Assumed mirrored B 4x16 layout: lane l -> N=l&15, Kbase=(l>>4)*2.
  // Per-lane values coincide, so one set serves as B (H-blur) and A (V-blur).
  const int wv   = tid >> 5;        // wave id 0..7 (uniform per wave)
  const int lane = tid & 31;
  const int qm   = lane & 15;       // M (A) / N (B) index
  const int hf   = lane >> 4;       // half-wave -> Kbase = 2*hf
  v2f wfrag[5];
#pragma unroll
  for (int j = 0; j < 5; ++j) {
    wfrag[j].x = band_w(4 * j + 2 * hf     - qm);
    wfrag[j].y = band_w(4 * j + 2 * hf + 1 - qm);
  }
#endif

  float gxs[16], gys[16];   // per-thread gx_sum/gy_sum for 16 center pixels

  for (int c = 0; c < 3; ++c) {
    const float* src = img + (((size_t)b * 3 + c) << 20);

    // ---- stage 0: stage 72x72 input tile into LDS (zero-padded) ----------
    if (fast) {
#if HAVE_TDM
      const int r0 = (int)wave * ROWS_PER_WAVE;            // this wave's rows
      const int gy0 = y0 - APRON + r0;
      tdm_load_rows(src + (size_t)gy0 * IMG + (x0 - APRON),
                    (unsigned)(uintptr_t)&sIn[r0 * IN_W],
                    ROWS_PER_WAVE,
                    (unsigned)(IMG - (x0 - APRON)),        // remaining x extent
                    (unsigned)(IMG - gy0));                // remaining y extent
      wait_tensor0();
#else
      for (int i = tid; i < IN_W * (IN_W / 4); i += NTHR) {
        int ly = i / (IN_W / 4), vx = i % (IN_W / 4);
        async_ld_b128(src + (size_t)(y0 - APRON + ly) * IMG + (x0 - APRON) + vx * 4,
                      &sIn[ly * IN_W + vx * 4]);
      }
      wait_async0();
#endif
    } else {
      for (int i = tid; i < IN_W * IN_W; i += NTHR) {
        int ly = i / IN_W, lx = i % IN_W;
        int gy = y0 - APRON + ly, gx = x0 - APRON + lx;
        if ((unsigned)gy < IMG && (unsigned)gx < IMG)
          async_ld_b32(src + (size_t)gy * IMG + gx, &sIn[i]);
        else
          sIn[i] = 0.0f;                 // zero-pad (disjoint addresses)
      }
      wait_async0();
    }
    __syncthreads();

    float* sVb = sIn;   // V-blur output aliases the (dead) raw-input buffer

#if HAVE_WMMA4
    // ---- stage 1 (WMMA): Hb[72][68], D = In(16x20) x Wband ----------------
    // Block grid: 5 row-blocks {0,16,32,48,56} x 5 col-blocks {0,16,32,48,52}
    // (overlapping edge blocks rewrite identical values). Wave-uniform loop
    // keeps EXEC all-ones as WMMA requires.
    for (int blk = wv; blk < 25; blk += 8) {
      int rb = blk / 5, cb = blk % 5;
      int yb = rb * 16; yb = (yb > 56) ? 56 : yb;
      int xb = cb * 16; xb = (xb > 52) ? 52 : xb;
      v8f d = {};
#pragma unroll
      for (int j = 0; j < 5; ++j) {
        v2f a = *(const v2f*)&sIn[(yb + qm) * IN_W + xb + 4 * j + 2 * hf];
        d = __builtin_amdgcn_wmma_f32_16x16x4_f32(false, a, false, wfrag[j],
                                                  (short)0, d, false, false);
      }
#pragma unroll
      for (int r = 0; r < 8; ++r)      // D: VGPR r, lane -> M=r+8*hf, N=qm
        sHb[(yb + r + 8 * hf) * HB_W + xb + qm] = d[r];
    }
    __syncthreads();

    // ---- stage 2 (WMMA): Vb[68][68], D = Wband^T(16x20) x Hb --------------
    for (int blk = wv; blk < 25; blk += 8) {
      int rb = blk / 5, cb = blk % 5;
      int yb = rb * 16; yb = (yb > 52) ? 52 : yb;
      int xb = cb * 16; xb = (xb > 52) ? 52 : xb;
      v8f d = {};
#pragma unroll
      for (int j = 0; j < 5; ++j) {
        v2f bf;
        bf.x = sHb[(yb + 4 * j + 2 * hf    ) * HB_W + xb + qm];
        bf.y = sHb[(yb + 4 * j + 2 * hf + 1) * HB_W + xb + qm];
        d = __builtin_amdgcn_wmma_f32_16x16x4_f32(false, wfrag[j], false, bf,
                                                  (short)0, d, false, false);
      }
#pragma unroll
      for (int r = 0; r < 8; ++r) {
        int vy = yb + r + 8 * hf, vx = xb + qm;
        int iy = y0 - 2 + vy, ix = x0 - 2 + vx;
        float v = d[r];
        // Reference zero-pads *blurred* before Sobel.
        if ((unsigned)iy >= IMG || (unsigned)ix >= IMG) v = 0.0f;
        sVb[vy * VB_W + vx] = v;
      }
    }
    __syncthreads();
#else
    // ---- stage 1 (VALU fallback): horizontal 5-tap blur -------------------
    for (int i = tid; i < IN_W * HB_W; i += NTHR) {
      int ly = i / HB_W, lx = i % HB_W;
      const float* r = &sIn[ly * IN_W + lx];
      sHb[i] = G0 * (r[0] + r[4]) + G1 * (r[1] + r[3]) + r[2];
    }
    __syncthreads();
    // ---- stage 2 (VALU fallback): vertical blur, zero outside image -------
    for (int i = tid; i < VB_W * VB_W; i += NTHR) {
      int vy = i / VB_W, vx = i % VB_W;
      float v = G0 * (sHb[vy * HB_W + vx] + sHb[(vy + 4) * HB_W + vx])
              + G1 * (sHb[(vy + 1) * HB_W + vx] + sHb[(vy + 3) * HB_W + vx])
              +       sHb[(vy + 2) * HB_W + vx];
      int iy = y0 - 2 + vy, ix = x0 - 2 + vx;
      if ((unsigned)iy >= IMG || (unsigned)ix >= IMG) v = 0.0f;
      sVb[vy * VB_W + vx] = v;
    }
    __syncthreads();
#endif

    // ---- stage 3: sobel + magnitude on 66x66 ring, accumulate sGr ---------
    // Reference zero-pads grad_mag before the directional filters.
    for (int i = tid; i < GR_W * GR_W; i += NTHR) {
      int ry = i / GR_W, rx = i % GR_W;          // center = Vb(ry+1, rx+1)
      const float* v0 = &sVb[(ry    ) * VB_W + rx];
      const float* v1 = &sVb[(ry + 1) * VB_W + rx];
      const float* v2 = &sVb[(ry + 2) * VB_W + rx];
      float gx = (v0[0] - v0[2]) + 2.0f * (v1[0] - v1[2]) + (v2[0] - v2[2]);
      float gy = (v0[0] + 2.0f * v0[1] + v0[2]) - (v2[0] + 2.0f * v2[1] + v2[2]);
      float m  = sqrtf(gx * gx + gy * gy + 1e-8f);
      int iy = y0 - 1 + ry, ix = x0 - 1 + rx;
      if ((unsigned)iy >= IMG || (unsigned)ix >= IMG) m = 0.0f;
      if (c == 0) sGr[i] = m; else sGr[i] += m;
    }

    // ---- stage 4: accumulate gx_sum/gy_sum for the 64x64 center in VGPRs --
#pragma unroll
    for (int k = 0; k < 16; ++k) {
      int i = tid + k * NTHR;                    // 0..4095
      int py = i >> 6, px = i & 63;              // Vb center = (py+2, px+2)
      const float* u0 = &sVb[(py + 1) * VB_W + (px + 1)];
      const float* u1 = u0 + VB_W;
      const float* u2 = u1 + VB_W;
      float gx = (u0[0] - u0[2]) + 2.0f * (u1[0] - u1[2]) + (u2[0] - u2[2]);
      float gy = (u0[0] + 2.0f * u0[1] + u0[2]) - (u2[0] + 2.0f * u2[1] + u2[2]);
      if (c == 0) { gxs[k] = gx; gys[k] = gy; }
      else        { gxs[k] += gx; gys[k] += gy; }
    }
    __syncthreads();   // sGr complete / protect sIn+sHb before next channel
  }

  // ---- NMS: octant select (exact tan(22.5) boundaries ~ round(atan2/45)) --
  const unsigned DYP = 0x00012221u;   // dy+1 per octant 0..7 (packed nibbles)
  const unsigned DXP = 0x21000122u;   // dx+1 per octant 0..7
#pragma unroll
  for (int k = 0; k < 16; ++k) {
    int i  = tid + k * NTHR;
    int py = i >> 6, px = i & 63;
    float m  = sGr[(py + 1) * GR_W + (px + 1)];
    float gx = gxs[k], gy = gys[k];
    float ax = fabsf(gx), ay = fabsf(gy);
    int o;
    if (ay <= TAN22 * ax)      o = (gx >= 0.0f) ? 4 : 0;   // horizontal grad
    else if (ax < TAN22 * ay)  o = (gy >= 0.0f) ? 6 : 2;   // vertical grad
    else if (gx >= 0.0f)       o = (gy >= 0.0f) ? 5 : 3;   // diagonals
    else                       o = (gy >= 0.0f) ? 7 : 1;
    int dy = (int)((DYP >> (o * 4)) & 7u) - 1;
    int dx = (int)((DXP >> (o * 4)) & 7u) - 1;
    int ci = (py + 1) * GR_W + (px + 1);
    float pos = m - sGr[ci + dy * GR_W + dx];   // dirs[idx_pos] * grad
    float neg = m - sGr[ci - dy * GR_W - dx];   // dirs[idx_neg] * grad
    float r = (fminf(pos, neg) > 0.0f && m >= 1.0f) ? m : 0.0f;
    int iy = y0 + py, ix = x0 + px;
    if (iy == 0 || iy == IMG - 1 || ix == 0 || ix == IMG - 1) r = 0.0f;
    out[((size_t)b << 20) + (size_t)iy * IMG + ix] = r;
  }
}

extern "C" void kernel_launch(void* const* d_in, const int* in_sizes, int n_in,
                              void* d_out, int out_size, void* d_ws, size_t ws_size,
                              hipStream_t stream) {
  const float* img = (const float*)d_in[0];
  float* out = (float*)d_out;
  int batches = in_sizes[0] / (3 * IMG * IMG);   // 8 for the reference shapes
  dim3 grid(IMG / TILE, IMG / TILE, batches);
  canny_fused_kernel<<<grid, dim3(NTHR), 0, stream>>>(img, out);
}